// AdaptiveKalmanFilterNN2_69329362092186
// MI455X (gfx1250) — compile-verified
//
#include <hip/hip_runtime.h>
#include <hip/hip_bf16.h>
#include <stddef.h>

#define TT 1024
#define BB 256
#define DD 64
#define AD 32
#define KF_EPS 1e-6f

typedef float v2f __attribute__((ext_vector_type(2)));
typedef float v8f __attribute__((ext_vector_type(8)));

// ---------------------------------------------------------------------------
// f32 WMMA 16x16x4: D = A(16x4) x B(4x16) + C(16x16)
// ---------------------------------------------------------------------------
__device__ __forceinline__ v8f wmma4(v2f a, v2f b, v8f c) {
    return __builtin_amdgcn_wmma_f32_16x16x4_f32(
        false, a, false, b, (short)0, c, false, false);
}

// A-fragment (16x4, rows r0..r0+15, K-block kb) from row-major [row][k], row
// stride rs.  Lane layout (ISA 7.12.2): lanes 0-15 hold K=0,1 ; lanes 16-31
// hold K=2,3 ; M = lane&15.  => contiguous float2 per lane.
// B-fragment (4x16) of a transposed weight W[n][k] has the identical access
// pattern with r0 = column-tile base, so one loader serves both.
__device__ __forceinline__ v2f fragAB(const float* base, int rs, int r0,
                                      int kb, int half, int l15) {
    const float* p = base + (size_t)(r0 + l15) * rs + (kb << 2) + (half << 1);
    return v2f{p[0], p[1]};
}

// ---------------------------------------------------------------------------
// Kernel A1: Q = LQ*LQ^T, R = LR*LR^T, per-step reset flags, state-source idx
// ---------------------------------------------------------------------------
__global__ __launch_bounds__(256) void kf_setup(
    const float* __restrict__ LQ, const float* __restrict__ LR,
    const unsigned char* __restrict__ is_init,   // jax bool -> 1 byte/elem
    float* __restrict__ Qg, float* __restrict__ Rg,
    int* __restrict__ reset, int* __restrict__ src) {
    int tid = threadIdx.x;
    for (int idx = tid; idx < DD * DD; idx += 256) {
        int i = idx >> 6, j = idx & 63;
        float sq = 0.f, sr = 0.f;
        for (int k = 0; k < DD; ++k) {
            sq += LQ[i * DD + k] * LQ[j * DD + k];
            sr += LR[i * DD + k] * LR[j * DD + k];
        }
        Qg[idx] = sq; Rg[idx] = sr;
    }
    for (int t = tid; t < TT; t += 256) {
        const unsigned char* p = is_init + (size_t)t * BB;
        int any = 0;
        for (int b = 0; b < BB; ++b) any |= p[b];
        reset[t] = any ? 1 : 0;
    }
    __syncthreads();
    if (tid == 0) {
        int last = -1;
        for (int t = 0; t < TT; ++t) { if (reset[t]) last = t; src[t] = last; }
    }
}

// ---------------------------------------------------------------------------
// Kernel A2: per reset-segment serial Kalman-gain chain.
// Block t activates only at segment starts; walks until the next reset.
// Gauss-Jordan on [S | HP^T] directly yields K^T (S symmetric + eps*I).
// ---------------------------------------------------------------------------
__global__ __launch_bounds__(512) void kf_gains(
    const float* __restrict__ Am, const float* __restrict__ Hm,
    const float* __restrict__ Qg, const float* __restrict__ Rg,
    const int* __restrict__ reset, float* __restrict__ K_all) {
    const int t0 = blockIdx.x;
    if (t0 > 0 && reset[t0] == 0) return;   // not a segment start

    __shared__ float P [DD][DD];
    __shared__ float Pp[DD][DD];
    __shared__ float HP[DD][DD];
    __shared__ float Km[DD][DD];
    __shared__ float Lf[DD][DD];
    __shared__ float T1[DD][DD];
    __shared__ float aug[DD][130];          // [S | rhs], padded

    const int tid = threadIdx.x;
    for (int idx = tid; idx < DD * DD; idx += 512) {
        int i = idx >> 6, j = idx & 63;
        P[i][j] = (i == j) ? 1.f : 0.f;     // reset / initial P = I
    }
    __syncthreads();

    for (int t = t0; t < TT; ++t) {
        if (t > t0 && reset[t]) break;      // next segment takes over
        // T1 = P @ A^T
        for (int idx = tid; idx < DD * DD; idx += 512) {
            int i = idx >> 6, j = idx & 63; float s = 0.f;
            for (int k = 0; k < DD; ++k) s += P[i][k] * Am[j * DD + k];
            T1[i][j] = s;
        }
        __syncthreads();
        // Pp = A @ T1 + Q
        for (int idx = tid; idx < DD * DD; idx += 512) {
            int i = idx >> 6, j = idx & 63; float s = 0.f;
            for (int k = 0; k < DD; ++k) s += Am[i * DD + k] * T1[k][j];
            Pp[i][j] = s + Qg[idx];
        }
        __syncthreads();
        // HP = Pp @ H^T
        for (int idx = tid; idx < DD * DD; idx += 512) {
            int i = idx >> 6, j = idx & 63; float s = 0.f;
            for (int k = 0; k < DD; ++k) s += Pp[i][k] * Hm[j * DD + k];
            HP[i][j] = s;
        }
        __syncthreads();
        // aug = [ H@HP + R + eps*I | HP^T ]
        for (int idx = tid; idx < DD * DD; idx += 512) {
            int i = idx >> 6, j = idx & 63; float s = 0.f;
            for (int k = 0; k < DD; ++k) s += Hm[i * DD + k] * HP[k][j];
            aug[i][j] = s + Rg[idx] + ((i == j) ? KF_EPS : 0.f);
            aug[i][64 + j] = HP[j][i];
        }
        __syncthreads();
        // Gauss-Jordan (no pivoting: S is eps-regularized SPD)
        for (int p = 0; p < DD; ++p) {
            float pv = aug[p][p];
            __syncthreads();
            if (tid < 128) aug[p][tid] *= (1.0f / pv);
            __syncthreads();
            int i = tid >> 3;
            int c0 = (tid & 7) << 4;
            float f = aug[i][p];
            __syncthreads();
            if (i != p)
                for (int c = c0; c < c0 + 16; ++c) aug[i][c] -= f * aug[p][c];
            __syncthreads();
        }
        // K = (inv(S) @ HP^T)^T ; store row-major for the WMMA B-operand
        for (int idx = tid; idx < DD * DD; idx += 512) {
            int i = idx >> 6, j = idx & 63;
            float v = aug[j][64 + i];
            Km[i][j] = v;
            K_all[(size_t)t * DD * DD + idx] = v;
        }
        __syncthreads();
        // Lf = I - K @ H
        for (int idx = tid; idx < DD * DD; idx += 512) {
            int i = idx >> 6, j = idx & 63; float s = 0.f;
            for (int k = 0; k < DD; ++k) s += Km[i][k] * Hm[k * DD + j];
            Lf[i][j] = ((i == j) ? 1.f : 0.f) - s;
        }
        __syncthreads();
        // T1 = Lf @ Pp
        for (int idx = tid; idx < DD * DD; idx += 512) {
            int i = idx >> 6, j = idx & 63; float s = 0.f;
            for (int k = 0; k < DD; ++k) s += Lf[i][k] * Pp[k][j];
            T1[i][j] = s;
        }
        __syncthreads();
        // P = T1 @ Lf^T
        for (int idx = tid; idx < DD * DD; idx += 512) {
            int i = idx >> 6, j = idx & 63; float s = 0.f;
            for (int k = 0; k < DD; ++k) s += T1[i][k] * Lf[j][k];
            P[i][j] = s;
        }
        __syncthreads();
        // Pp := K @ R   (Pp dead, reuse as temp)
        for (int idx = tid; idx < DD * DD; idx += 512) {
            int i = idx >> 6, j = idx & 63; float s = 0.f;
            for (int k = 0; k < DD; ++k) s += Km[i][k] * Rg[k * DD + j];
            Pp[i][j] = s;
        }
        __syncthreads();
        // P += Pp @ K^T   (Joseph form)
        for (int idx = tid; idx < DD * DD; idx += 512) {
            int i = idx >> 6, j = idx & 63; float s = 0.f;
            for (int k = 0; k < DD; ++k) s += Pp[i][k] * Km[j][k];
            P[i][j] += s;
        }
        __syncthreads();
    }
}

// ---------------------------------------------------------------------------
// Kernel B: batched measurement update, three chained f32 WMMA GEMMs.
// grid = (BATCH/64, T); block = 128 (4 waves, 16 batch rows each).
// ---------------------------------------------------------------------------
__global__ __launch_bounds__(128) void kf_update(
    const float* __restrict__ se, const float* __restrict__ pa,
    const float* __restrict__ ca, const float* __restrict__ ob,
    const float* __restrict__ Am, const float* __restrict__ Bm,
    const float* __restrict__ Hm, const float* __restrict__ K_all,
    const int* __restrict__ src,
    float* __restrict__ out_upd, float* __restrict__ out_ca,
    float* __restrict__ out_err) {
    const int t     = blockIdx.y;
    const int btile = blockIdx.x;           // 64 batch rows per block
    const int tid   = threadIdx.x;
    const int w     = tid >> 5;
    const int lane  = tid & 31;
    const int half  = lane >> 4;
    const int l15   = lane & 15;
    const int m0    = btile * 64 + w * 16;  // this wave's batch-row base

    __shared__ float sPred[4][16][68];      // per-wave pred strip (16x64)
    __shared__ float sInn [4][16][68];      // per-wave innov strip

    const float* ob_t = ob + (size_t)t * BB * DD;
    const float* pa_t = pa + (size_t)t * BB * AD;
    const float* Kt   = K_all + (size_t)t * DD * DD;
    const int s       = src[t];
    const float* st_t = (s >= 0) ? (ob + (size_t)s * BB * DD) : se; // se slice 0

    // passthrough: current_action
    {
        const float* ca_t = ca + (size_t)t * BB * AD + btile * 64 * AD;
        float*       oc   = out_ca + (size_t)t * BB * AD + btile * 64 * AD;
        for (int i = tid; i < 64 * AD; i += 128) oc[i] = ca_t[i];
    }

    // ---- Phase 1: pred = state_src @ A^T + pa @ B^T -----------------------
    v8f acc[4];
#pragma unroll
    for (int nb = 0; nb < 4; ++nb) {
        v8f c = {0.f, 0.f, 0.f, 0.f, 0.f, 0.f, 0.f, 0.f};
#pragma unroll
        for (int kb = 0; kb < 16; ++kb)     // K = 64
            c = wmma4(fragAB(st_t, DD, m0, kb, half, l15),
                      fragAB(Am,   DD, nb * 16, kb, half, l15), c);
#pragma unroll
        for (int kb = 0; kb < 8; ++kb)      // K = 32
            c = wmma4(fragAB(pa_t, AD, m0, kb, half, l15),
                      fragAB(Bm,   AD, nb * 16, kb, half, l15), c);
        acc[nb] = c;
#pragma unroll
        for (int r = 0; r < 8; ++r)         // C-layout -> LDS strip
            sPred[w][r + 8 * half][nb * 16 + l15] = c[r];
    }
    __syncthreads();

    // ---- Phase 2: innov = ob - pred @ H^T ---------------------------------
#pragma unroll
    for (int nb = 0; nb < 4; ++nb) {
        v8f c = {0.f, 0.f, 0.f, 0.f, 0.f, 0.f, 0.f, 0.f};
#pragma unroll
        for (int kb = 0; kb < 16; ++kb) {
            const float* p = &sPred[w][l15][(kb << 2) + (half << 1)];
            v2f a = {p[0], p[1]};
            c = wmma4(a, fragAB(Hm, DD, nb * 16, kb, half, l15), c);
        }
#pragma unroll
        for (int r = 0; r < 8; ++r) {
            int row = m0 + r + 8 * half;
            int col = nb * 16 + l15;
            float inn = ob_t[(size_t)row * DD + col] - c[r];
            sInn[w][r + 8 * half][col] = inn;
            out_err[(size_t)t * BB * DD + (size_t)row * DD + col] = inn;
        }
    }
    __syncthreads();

    // ---- Phase 3: upd = pred + innov @ K^T --------------------------------
#pragma unroll
    for (int nb = 0; nb < 4; ++nb) {
        v8f c = acc[nb];
#pragma unroll
        for (int kb = 0; kb < 16; ++kb) {
            const float* p = &sInn[w][l15][(kb << 2) + (half << 1)];
            v2f a = {p[0], p[1]};
            c = wmma4(a, fragAB(Kt, DD, nb * 16, kb, half, l15), c);
        }
#pragma unroll
        for (int r = 0; r < 8; ++r) {
            int row = m0 + r + 8 * half;
            int col = nb * 16 + l15;
            out_upd[(size_t)t * BB * DD + (size_t)row * DD + col] = c[r];
        }
    }
}

// ---------------------------------------------------------------------------
extern "C" void kernel_launch(void* const* d_in, const int* in_sizes, int n_in,
                              void* d_out, int out_size, void* d_ws,
                              size_t ws_size, hipStream_t stream) {
    (void)in_sizes; (void)n_in; (void)out_size; (void)ws_size;
    const float* se = (const float*)d_in[0];               // (T,B,D)
    const float* pa = (const float*)d_in[1];               // (T,B,AD)
    const float* ca = (const float*)d_in[2];               // (T,B,AD)
    const float* ob = (const float*)d_in[3];               // (T,B,D)
    const unsigned char* ii = (const unsigned char*)d_in[4]; // (T,B) bool
    const float* Am = (const float*)d_in[5];               // (D,D)
    const float* Bm = (const float*)d_in[6];               // (D,AD)
    const float* Hm = (const float*)d_in[7];               // (D,D)
    const float* LQ = (const float*)d_in[8];               // (D,D)
    const float* LR = (const float*)d_in[9];               // (D,D)

    float* out_upd = (float*)d_out;
    float* out_ca  = out_upd + (size_t)TT * BB * DD;
    float* out_err = out_ca  + (size_t)TT * BB * AD;

    float* wsf   = (float*)d_ws;
    float* K_all = wsf;                         // T*64*64 floats (16 MB)
    float* Qg    = wsf + (size_t)TT * DD * DD;
    float* Rg    = Qg + DD * DD;
    int*   reset = (int*)(Rg + DD * DD);
    int*   src   = reset + TT;

    kf_setup<<<1, 256, 0, stream>>>(LQ, LR, ii, Qg, Rg, reset, src);
    kf_gains<<<TT, 512, 0, stream>>>(Am, Hm, Qg, Rg, reset, K_all);
    kf_update<<<dim3(BB / 64, TT), 128, 0, stream>>>(
        se, pa, ca, ob, Am, Bm, Hm, K_all, src, out_upd, out_ca, out_err);
}